// DeepGCN_Edge_81123342287180
// MI455X (gfx1250) — compile-verified
//
#include <hip/hip_runtime.h>

typedef float v2f __attribute__((ext_vector_type(2)));
typedef float v8f __attribute__((ext_vector_type(8)));

#define DIM      128
#define TWO_D    256
#define TILE_E   128          // edges per workgroup tile
#define HSTRIDE  260          // padded LDS row stride (floats): conflict-free b64 loads
#define LN_EPS   1e-5f

// LDS layout (floats):
//   [WT_OFF)   Wt[n*HSTRIDE + k] = W[k*128 + n]   (128 rows x 256, padded)
//   [H_OFF)    h[e*HSTRIDE + f]  (128 rows x 256, padded), normalized+relu
//   gamma[256], beta[256], bias[128]
#define WT_OFF    0
#define H_OFF     (128 * HSTRIDE)
#define G_OFF     (2 * 128 * HSTRIDE)
#define BETA_OFF  (G_OFF + TWO_D)
#define BIAS_OFF  (BETA_OFF + TWO_D)
#define SMEM_FLOATS (BIAS_OFF + DIM)

__global__ __launch_bounds__(256)
void deepgcn_edge_fused(const float* __restrict__ x,
                        const int*   __restrict__ edge_index,
                        const float* __restrict__ edge_attr,
                        const float* __restrict__ ln_gamma,
                        const float* __restrict__ ln_beta,
                        const float* __restrict__ W,
                        const float* __restrict__ bias,
                        float* __restrict__ out,
                        int E)
{
    extern __shared__ float smem[];
    float* wt_lds = smem + WT_OFF;
    float* h_lds  = smem + H_OFF;
    float* g_lds  = smem + G_OFF;
    float* be_lds = smem + BETA_OFF;
    float* bi_lds = smem + BIAS_OFF;

    const int t = threadIdx.x;

    // One-time preload: W transposed into LDS, plus gamma/beta/bias.
    for (int idx = t; idx < TWO_D * DIM; idx += 256) {
        const int k = idx >> 7;          // 0..255
        const int n = idx & (DIM - 1);   // 0..127
        wt_lds[n * HSTRIDE + k] = W[idx];
    }
    if (t < TWO_D) { g_lds[t] = ln_gamma[t]; be_lds[t] = ln_beta[t]; }
    if (t < DIM)   { bi_lds[t] = bias[t]; }

    const int lane  = t & 31;
    const int wave  = t >> 5;       // 0..7
    const int e_loc = t >> 1;       // 0..127 (2 threads per edge)
    const int half  = t & 1;        // 0 = src row, 1 = dst row
    const int m     = lane & 15;    // M (edges) / N (cols) index within fragment
    const int kg    = lane >> 4;    // lane-group: 0 or 1

    const int num_tiles = (E + TILE_E - 1) / TILE_E;

    for (int tile = blockIdx.x; tile < num_tiles; tile += (int)gridDim.x) {
        __syncthreads();   // covers preload on first iter; protects h_lds reuse after

        // ---------- Stage 1: gather + LayerNorm + ReLU -> h tile in LDS ----------
        const int e_g = tile * TILE_E + e_loc;
        int node = 0;
        if (e_g < E) node = edge_index[half * E + e_g];
        const float4* __restrict__ xr = (const float4*)(x + (size_t)node * DIM);

        float4 v[32];
        float s = 0.f, sq = 0.f;
        #pragma unroll
        for (int j = 0; j < 32; ++j) {
            float4 u = xr[j];
            v[j] = u;
            s  += u.x + u.y + u.z + u.w;
            sq += u.x * u.x + u.y * u.y + u.z * u.z + u.w * u.w;
        }
        // combine the two half-rows of this edge (partner lane = lane^1)
        s  += __shfl_xor(s,  1, 32);
        sq += __shfl_xor(sq, 1, 32);

        const float mu  = s  * (1.f / 256.f);
        const float var = sq * (1.f / 256.f) - mu * mu;
        const float rs  = rsqrtf(var + LN_EPS);

        float* hr = h_lds + e_loc * HSTRIDE + half * DIM;
        const float* gh = g_lds  + half * DIM;
        const float* bh = be_lds + half * DIM;
        #pragma unroll
        for (int j = 0; j < 32; ++j) {
            float4 u = v[j];
            float4 r;
            r.x = fmaxf((u.x - mu) * rs * gh[4*j + 0] + bh[4*j + 0], 0.f);
            r.y = fmaxf((u.y - mu) * rs * gh[4*j + 1] + bh[4*j + 1], 0.f);
            r.z = fmaxf((u.z - mu) * rs * gh[4*j + 2] + bh[4*j + 2], 0.f);
            r.w = fmaxf((u.w - mu) * rs * gh[4*j + 3] + bh[4*j + 3], 0.f);
            *(float4*)(hr + 4*j) = r;
        }
        __syncthreads();

        // ---------- Stage 2: WMMA GEMM: wave computes [16 edges x 128 cols] ----------
        // A fragment (16x4 f32): lane l -> row m = l&15; vgpr0/1 hold K = k0+2*kg, k0+2*kg+1
        const float* ha = h_lds + (wave * 16 + m) * HSTRIDE + kg * 2;

        v8f c[8] = {};
        for (int kt = 0; kt < 64; ++kt) {
            const int k0 = kt * 4;
            v2f a = *(const v2f*)(ha + k0);
            #pragma unroll
            for (int nt = 0; nt < 8; ++nt) {
                // B fragment (4x16 f32): lane l -> col n = nt*16 + (l&15);
                // vgpr0/1 hold rows K = k0+2*kg, k0+2*kg+1 (contiguous in transposed W)
                v2f b = *(const v2f*)(wt_lds + (nt * 16 + m) * HSTRIDE + k0 + kg * 2);
                c[nt] = __builtin_amdgcn_wmma_f32_16x16x4_f32(
                            false, a, false, b, (short)0, c[nt], false, false);
            }
        }

        // ---------- Stage 3: bias + residual + store ----------
        // C/D layout: VGPR i -> M = i (lanes 0-15), M = i+8 (lanes 16-31); N = lane&15
        #pragma unroll
        for (int nt = 0; nt < 8; ++nt) {
            const int n  = nt * 16 + m;
            const float bn = bi_lds[n];
            #pragma unroll
            for (int i = 0; i < 8; ++i) {
                const int ee = tile * TILE_E + wave * 16 + kg * 8 + i;
                if (ee < E) {
                    const size_t off = (size_t)ee * DIM + n;
                    out[off] = edge_attr[off] + c[nt][i] + bn;
                }
            }
        }
    }
}

extern "C" void kernel_launch(void* const* d_in, const int* in_sizes, int n_in,
                              void* d_out, int out_size, void* d_ws, size_t ws_size,
                              hipStream_t stream) {
    const float* x   = (const float*)d_in[0];
    const int*   ei  = (const int*)  d_in[1];
    const float* ea  = (const float*)d_in[2];
    const float* g   = (const float*)d_in[3];
    const float* be  = (const float*)d_in[4];
    const float* W   = (const float*)d_in[5];
    const float* b   = (const float*)d_in[6];
    float* out = (float*)d_out;

    const int E = in_sizes[2] / DIM;              // edge_attr is [E, 128]
    const int tiles = (E + TILE_E - 1) / TILE_E;
    const int grid  = tiles < 1024 ? tiles : 1024;
    const size_t shmem = (size_t)SMEM_FLOATS * sizeof(float);

    deepgcn_edge_fused<<<grid, 256, shmem, stream>>>(x, ei, ea, g, be, W, b, out, E);
}